// Decoder_16741782520502
// MI455X (gfx1250) — compile-verified
//
#include <hip/hip_runtime.h>
#include <math.h>

#define XSIZE 256
#define HN    129            // rfft half-width
#define FW    144            // padded half-width (multiple of 16; K-mult of 4)
#define BATCH 32
#define NPTS  200000

typedef float v2f __attribute__((ext_vector_type(2)));
typedef float v8f __attribute__((ext_vector_type(8)));

// ---- workspace layout (float offsets) ----
constexpr size_t SZ_IMG   = (size_t)BATCH * XSIZE * XSIZE;   // 2,097,152
constexpr size_t SZ_P     = (size_t)BATCH * XSIZE * FW;      // 1,179,648
constexpr size_t OFF_IMG  = 0;
constexpr size_t OFF_P0R  = OFF_IMG  + SZ_IMG;   // Fxr, later Gr
constexpr size_t OFF_P0I  = OFF_P0R  + SZ_P;     // Fxi, later Gi
constexpr size_t OFF_P1R  = OFF_P0I  + SZ_P;     // Fr -> Hr (in place)
constexpr size_t OFF_P1I  = OFF_P1R  + SZ_P;     // Fi -> Hi
constexpr size_t OFF_ROT  = OFF_P1I  + SZ_P;     // BATCH*12
constexpr size_t OFF_CCUT = OFF_ROT  + BATCH*12; // 256 x FW : cos, cols>=129 zero
constexpr size_t OFF_NSCUT= OFF_CCUT + XSIZE*FW; // 256 x FW : -sin
constexpr size_t OFF_C256 = OFF_NSCUT+ XSIZE*FW; // 256 x 256 cos
constexpr size_t OFF_S256 = OFF_C256 + XSIZE*XSIZE;
constexpr size_t OFF_NS256= OFF_S256 + XSIZE*XSIZE;
constexpr size_t OFF_ER   = OFF_NS256+ XSIZE*XSIZE; // FW x 256 irfft-x real basis
constexpr size_t OFF_EI   = OFF_ER   + FW*XSIZE;    // FW x 256 irfft-x imag basis

__device__ __forceinline__ v8f wmma4(v2f a, v2f b, v8f c) {
    return __builtin_amdgcn_wmma_f32_16x16x4_f32(false, a, false, b,
                                                 (short)0, c, false, false);
}

// ---------------- zero the accumulation images ----------------
__global__ void zero_img_kernel(float* __restrict__ img) {
    size_t i = (size_t)blockIdx.x * blockDim.x + threadIdx.x;
    if (i < SZ_IMG) img[i] = 0.0f;
}

// ---------------- 6D -> rotation rows (only rows 0,1 needed) ----------------
__global__ void rot6d_kernel(const float* __restrict__ al,
                             const float* __restrict__ sh,
                             float* __restrict__ rot) {
    int b = threadIdx.x;
    if (b >= BATCH) return;
    float a1x = al[b*6+0], a1y = al[b*6+1], a1z = al[b*6+2];
    float a2x = al[b*6+3], a2y = al[b*6+4], a2z = al[b*6+5];
    float n1 = rsqrtf(a1x*a1x + a1y*a1y + a1z*a1z);
    float b1x = a1x*n1, b1y = a1y*n1, b1z = a1z*n1;
    float d = b1x*a2x + b1y*a2y + b1z*a2z;
    float u2x = a2x - d*b1x, u2y = a2y - d*b1y, u2z = a2z - d*b1z;
    float n2 = rsqrtf(u2x*u2x + u2y*u2y + u2z*u2z);
    float* r = rot + b*12;
    r[0] = b1x;    r[1] = b1y;    r[2] = b1z;
    r[3] = u2x*n2; r[4] = u2y*n2; r[5] = u2z*n2;
    r[6] = 0.f;    r[7] = 0.f;    r[8] = 0.f;
    r[9] = sh[b*2+0]; r[10] = sh[b*2+1]; r[11] = 0.f;
}

// ---------------- DFT basis tables (exact angle reduction mod 256) ----------
__global__ void basis_kernel(float* __restrict__ ws) {
    const int t   = blockIdx.y;
    const int idx = blockIdx.x * blockDim.x + threadIdx.x;
    const float th    = 6.28318530717958647692f / 256.0f;
    const float scale = 1.0f / 65536.0f;   // 1/(256*256) irfft2 normalization
    if (t < 2) {                                   // Ccut / -Scut : [x][k], ld=FW
        if (idx >= XSIZE * FW) return;
        int x = idx / FW, k = idx % FW;
        float v = 0.0f;
        if (k < HN) {
            float ang = th * (float)((x * k) & 255);
            float s, c; sincosf(ang, &s, &c);
            v = (t == 0) ? c : -s;
        }
        ws[(t == 0 ? OFF_CCUT : OFF_NSCUT) + idx] = v;
    } else if (t < 5) {                            // C256 / S256 / -S256
        if (idx >= XSIZE * XSIZE) return;
        int i = idx >> 8, j = idx & 255;
        float ang = th * (float)((i * j) & 255);
        float s, c; sincosf(ang, &s, &c);
        ws[(t == 2 ? OFF_C256 : (t == 3 ? OFF_S256 : OFF_NS256)) + idx] =
            (t == 2) ? c : (t == 3 ? s : -s);
    } else {                                       // Er / Ei : [k][x], ld=256
        if (idx >= FW * XSIZE) return;
        int k = idx >> 8, x = idx & 255;
        float v = 0.0f;
        if (k < HN) {
            float w = (k == 0 || k == 128) ? 1.0f : 2.0f;
            float ang = th * (float)((k * x) & 255);
            float s, c; sincosf(ang, &s, &c);
            v = (t == 5) ? w * scale * c : -w * scale * s;
        }
        ws[(t == 5 ? OFF_ER : OFF_EI) + idx] = v;
    }
}

// ---------------- bilinear scatter with HW f32 atomics ----------------
__global__ void scatter_kernel(const float* __restrict__ coords,
                               const float* __restrict__ values,
                               const float* __restrict__ rot,
                               float* __restrict__ img) {
    const int b = blockIdx.y;
    const float* rb = rot + b * 12;
    const float r00 = rb[0], r01 = rb[1], r02 = rb[2];
    const float r10 = rb[3], r11 = rb[4], r12 = rb[5];
    const float sx = rb[9] + 128.0f, sy = rb[10] + 128.0f;
    float* im = img + (size_t)b * XSIZE * XSIZE;
    const int stride = gridDim.x * blockDim.x;
    for (int n = blockIdx.x * blockDim.x + threadIdx.x; n < NPTS; n += stride) {
        float cx = coords[3*n+0], cy = coords[3*n+1], cz = coords[3*n+2];
        float v  = values[n];
        float gx = r00*cx + r01*cy + r02*cz + sx;
        float gy = r10*cx + r11*cy + r12*cz + sy;
        float x0f = fminf(fmaxf(floorf(gx), 0.0f), 254.0f);
        float y0f = fminf(fmaxf(floorf(gy), 0.0f), 254.0f);
        int x0 = (int)x0f, y0 = (int)y0f;
        float fx = gx - x0f, fy = gy - y0f;
        float* p0 = &im[y0 * XSIZE + x0];
        __hip_atomic_fetch_add(p0,         v*(1.f-fx)*(1.f-fy), __ATOMIC_RELAXED, __HIP_MEMORY_SCOPE_AGENT);
        __hip_atomic_fetch_add(p0+1,       v*fx*(1.f-fy),       __ATOMIC_RELAXED, __HIP_MEMORY_SCOPE_AGENT);
        __hip_atomic_fetch_add(p0+XSIZE,   v*(1.f-fx)*fy,       __ATOMIC_RELAXED, __HIP_MEMORY_SCOPE_AGENT);
        __hip_atomic_fetch_add(p0+XSIZE+1, v*fx*fy,             __ATOMIC_RELAXED, __HIP_MEMORY_SCOPE_AGENT);
    }
}

// ---------------- pointwise gauss*ctf on Fr/Fi (in place) ----------------
__global__ void filter_kernel(const float* __restrict__ ctf,
                              float* __restrict__ hr, float* __restrict__ hi) {
    const int per = XSIZE * HN;
    int idx = blockIdx.x * blockDim.x + threadIdx.x;
    if (idx >= BATCH * per) return;
    int b = idx / per, r = idx % per;
    int ky = r / HN, k = r % HN;
    float fy = (float)(ky < 128 ? ky : ky - 256) * (1.0f / 256.0f);
    float fx = (float)k * (1.0f / 256.0f);
    const float c2 = -2.0f * 9.86960440108935862f; // -2*pi^2*sigma^2
    float filt = expf(c2 * (fx*fx + fy*fy)) * ctf[idx];
    size_t o = (size_t)b * XSIZE * FW + (size_t)ky * FW + k;
    hr[o] *= filt;
    hi[o] *= filt;
}

// =====================================================================
// Fused f32 WMMA GEMM.  M=256 fixed, N = gridDim.x*16, K = template.
// Block = 4 waves; wave w owns rows [w*64, w*64+64) as 4 16x16 subtiles.
// Both 16-wide B panels staged in LDS once, shared by all 4 waves.
// MODE 0 (shared-A, stage A):   D1 = A1@B1        ; D2 = A1@B2
// MODE 1 (complex, stages B/D): D1 = A1@B1 + A2@B2; D2 = A1@B2 - A2@B1
// MODE 2 (dual, stage E):       D1 = A1@B1 + A2@B2
// =====================================================================
template<int K, int MODE>
__global__ __launch_bounds__(128) void cgemm_wmma(
    const float* __restrict__ A1, const float* __restrict__ A2,
    const float* __restrict__ B1, const float* __restrict__ B2,
    float* __restrict__ D1, float* __restrict__ D2,
    int lda, int ldb, int ldd, long a_bs, long b_bs, long d_bs)
{
    __shared__ float ldsB[2 * K * 16];
    const int b   = blockIdx.y;
    const int tn  = blockIdx.x << 4;
    const int tid = threadIdx.y * 32 + threadIdx.x;

    // Cooperative staging of both K x 16 B panels into LDS (float4 rows).
    const float* gB1 = B1 + (size_t)b * b_bs + tn;
    const float* gB2 = B2 + (size_t)b * b_bs + tn;
    for (int i = tid; i < K * 4; i += 128) {
        int row = i >> 2, q = (i & 3) << 2;
        *(float4*)&ldsB[row * 16 + q] =
            *(const float4*)(gB1 + (size_t)row * ldb + q);
        *(float4*)&ldsB[K * 16 + row * 16 + q] =
            *(const float4*)(gB2 + (size_t)row * ldb + q);
    }
    __syncthreads();

    const int lane  = threadIdx.x;
    const int l15   = lane & 15;
    const int kOff  = (lane < 16) ? 0 : 2;     // A/B fragment K-halves
    const int waveM = threadIdx.y << 6;        // 64 rows per wave

    const float* pa1 = A1 + (size_t)b * a_bs + (size_t)(waveM + l15) * lda + kOff;
    const float* pa2 = A2 + (size_t)b * a_bs + (size_t)(waveM + l15) * lda + kOff;

    v8f accR[4] = {};
    v8f accI[4] = {};

#pragma unroll 2
    for (int kk = 0; kk < K; kk += 4) {
        const int kb = kk + kOff;
        v2f b1f, b2f;
        b1f.x = ldsB[kb * 16 + l15];
        b1f.y = ldsB[(kb + 1) * 16 + l15];
        b2f.x = ldsB[K * 16 + kb * 16 + l15];
        b2f.y = ldsB[K * 16 + (kb + 1) * 16 + l15];
#pragma unroll
        for (int st = 0; st < 4; ++st) {
            const size_t ao = (size_t)(st << 4) * lda + kk;
            v2f a1f = *(const v2f*)(pa1 + ao);
            if constexpr (MODE == 0) {
                accR[st] = wmma4(a1f, b1f, accR[st]);
                accI[st] = wmma4(a1f, b2f, accI[st]);
            } else if constexpr (MODE == 1) {
                v2f a2f = *(const v2f*)(pa2 + ao);
                v2f a2n; a2n.x = -a2f.x; a2n.y = -a2f.y;  // f32 WMMA has no A-neg
                accR[st] = wmma4(a1f, b1f, accR[st]);
                accR[st] = wmma4(a2f, b2f, accR[st]);
                accI[st] = wmma4(a1f, b2f, accI[st]);
                accI[st] = wmma4(a2n, b1f, accI[st]);
            } else {
                v2f a2f = *(const v2f*)(pa2 + ao);
                accR[st] = wmma4(a1f, b1f, accR[st]);
                accR[st] = wmma4(a2f, b2f, accR[st]);
            }
        }
    }

    // C/D layout: VGPR r -> row base+r (lanes 0-15) / base+8+r (lanes 16-31)
    const int rbase = (lane < 16) ? 0 : 8;
#pragma unroll
    for (int st = 0; st < 4; ++st) {
        const size_t rowo = (size_t)(waveM + (st << 4) + rbase) * ldd + tn + l15;
        float* d1 = D1 + (size_t)b * d_bs + rowo;
#pragma unroll
        for (int r = 0; r < 8; ++r) d1[(size_t)r * ldd] = accR[st][r];
        if constexpr (MODE != 2) {
            float* d2 = D2 + (size_t)b * d_bs + rowo;
#pragma unroll
            for (int r = 0; r < 8; ++r) d2[(size_t)r * ldd] = accI[st][r];
        }
    }
}

extern "C" void kernel_launch(void* const* d_in, const int* in_sizes, int n_in,
                              void* d_out, int out_size, void* d_ws, size_t ws_size,
                              hipStream_t stream) {
    const float* alignment = (const float*)d_in[0];
    const float* shifts    = (const float*)d_in[1];
    const float* coords    = (const float*)d_in[2];
    const float* values    = (const float*)d_in[3];
    const float* ctf       = (const float*)d_in[4];
    float* out = (float*)d_out;
    float* ws  = (float*)d_ws;

    zero_img_kernel<<<(SZ_IMG + 255) / 256, 256, 0, stream>>>(ws + OFF_IMG);
    rot6d_kernel<<<1, 32, 0, stream>>>(alignment, shifts, ws + OFF_ROT);
    basis_kernel<<<dim3(256, 7), 256, 0, stream>>>(ws);
    scatter_kernel<<<dim3(128, BATCH), 256, 0, stream>>>(coords, values,
                                                         ws + OFF_ROT, ws + OFF_IMG);

    const dim3 blk(32, 4);
    const long PB = (long)XSIZE * FW;      // 36864 per-batch F stride
    const long IB = (long)XSIZE * XSIZE;   // 65536 per-batch img stride

    // Stage A (rfft-x): Fxr = img@Ccut ; Fxi = img@(-Scut)       N=144
    cgemm_wmma<256, 0><<<dim3(FW / 16, BATCH), blk, 0, stream>>>(
        ws + OFF_IMG, ws + OFF_IMG, ws + OFF_CCUT, ws + OFF_NSCUT,
        ws + OFF_P0R, ws + OFF_P0I, 256, FW, FW, IB, 0, PB);

    // Stage B (fft-y): Fr = C@Fxr + S@Fxi ; Fi = C@Fxi - S@Fxr   N=144
    cgemm_wmma<256, 1><<<dim3(FW / 16, BATCH), blk, 0, stream>>>(
        ws + OFF_C256, ws + OFF_S256, ws + OFF_P0R, ws + OFF_P0I,
        ws + OFF_P1R, ws + OFF_P1I, 256, FW, FW, 0, PB, PB);

    // Stage C: H = F * gauss * ctf (in place)
    filter_kernel<<<(BATCH * XSIZE * HN + 255) / 256, 256, 0, stream>>>(
        ctf, ws + OFF_P1R, ws + OFF_P1I);

    // Stage D (ifft-y): Gr = C@Hr - S@Hi ; Gi = C@Hi + S@Hr  (A2 = -S)
    cgemm_wmma<256, 1><<<dim3(FW / 16, BATCH), blk, 0, stream>>>(
        ws + OFF_C256, ws + OFF_NS256, ws + OFF_P1R, ws + OFF_P1I,
        ws + OFF_P0R, ws + OFF_P0I, 256, FW, FW, 0, PB, PB);

    // Stage E (irfft-x): out = Gr@Er + Gi@Ei                    N=256, K=144
    cgemm_wmma<144, 2><<<dim3(XSIZE / 16, BATCH), blk, 0, stream>>>(
        ws + OFF_P0R, ws + OFF_P0I, ws + OFF_ER, ws + OFF_EI,
        out, out, FW, 256, 256, PB, 0, IB);

    (void)in_sizes; (void)n_in; (void)out_size; (void)ws_size;
}